// HeteroGNN_21105469292712
// MI455X (gfx1250) — compile-verified
//
#include <hip/hip_runtime.h>
#include <hip/hip_bf16.h>
#include <math.h>

// ---------------------------------------------------------------------------
// 2-layer bipartite GATConv (user -> badge), PyG semantics, fp32.
// Dense projections via v_wmma_f32_16x16x4_f32 (K=16 as 4 chained 16x16x4);
// edge softmax + scatter via L2 float atomics (accumulators 3.2MB << 192MB L2).
// ---------------------------------------------------------------------------

typedef __attribute__((ext_vector_type(2))) float v2f;
typedef __attribute__((ext_vector_type(8))) float v8f;

#define NEG_SLOPE 0.2f

// ---- ordered-uint encoding for float atomic max --------------------------
__device__ __forceinline__ unsigned f2ord(float f) {
    unsigned u = __float_as_uint(f);
    return (u & 0x80000000u) ? ~u : (u | 0x80000000u);
}
__device__ __forceinline__ float ord2f(unsigned o) {
    return (o & 0x80000000u) ? __uint_as_float(o ^ 0x80000000u)
                             : __uint_as_float(~o);
}

// ---------------------------------------------------------------------------
// H[nrows, COLS] = X[nrows,16] @ W[16,COLS]   (COLS <= 16, B zero-padded)
// alpha[nrows]   = H . att                     (att zero-padded past COLS)
// One wave32 per 16-row tile; K=16 as 4 chained WMMA f32 16x16x4.
// COLS is compile-time: COLS==16 path has no EXEC-divergent guards.
// ---------------------------------------------------------------------------
template <int COLS>
__global__ void gemm16_wmma(const float* __restrict__ X,
                            const float* __restrict__ W,
                            const float* __restrict__ att,
                            float* __restrict__ H,
                            float* __restrict__ alpha,
                            int nrows) {
    const int lane = threadIdx.x & 31;
    const int wave = threadIdx.x >> 5;
    const int tile = blockIdx.x * (blockDim.x >> 5) + wave;
    const int row0 = tile * 16;
    if (row0 >= nrows) return;          // wave-uniform guard (EXEC stays full)

    const int hi = lane >> 4;           // half-wave select (0/1)
    const int lm = lane & 15;           // M for A, N for B/C/D

    // B fragments for the 4 K-chunks: lane holds W[k0+2*hi + {0,1}][lm]
    v2f bfrag[4];
#pragma unroll
    for (int kc = 0; kc < 4; ++kc) {
        const int k = kc * 4 + 2 * hi;
        float b0, b1;
        if (COLS == 16 || lm < COLS) {
            b0 = W[k * COLS + lm];
            b1 = W[(k + 1) * COLS + lm];
        } else {
            b0 = 0.f;
            b1 = 0.f;
        }
        bfrag[kc][0] = b0;
        bfrag[kc][1] = b1;
    }

    v8f c = {};
    const float* xrow = X + (size_t)(row0 + lm) * 16;  // A: lane's row = lm
#pragma unroll
    for (int kc = 0; kc < 4; ++kc) {
        const int k = kc * 4 + 2 * hi;
        v2f a;
        a[0] = xrow[k];
        a[1] = xrow[k + 1];
        // (neg_a, A, neg_b, B, c_mod, C, reuse_a, reuse_b)
        c = __builtin_amdgcn_wmma_f32_16x16x4_f32(
                false, a, false, bfrag[kc], (short)0, c, false, false);
    }

    // c[r] = H[row0 + r + 8*hi][lm]
    const float av = (COLS == 16 || lm < COLS) ? att[lm] : 0.f;
    float rd[8];
#pragma unroll
    for (int r = 0; r < 8; ++r) {
        if (COLS == 16 || lm < COLS)
            H[(size_t)(row0 + r + 8 * hi) * COLS + lm] = c[r];
        rd[r] = c[r] * av;
    }
    // reduce over the 16 lanes of each half-wave (columns of each row)
#pragma unroll
    for (int off = 1; off <= 8; off <<= 1) {
#pragma unroll
        for (int r = 0; r < 8; ++r)
            rd[r] += __shfl_xor(rd[r], off, 32);
    }
    if (lm == 0) {
#pragma unroll
        for (int r = 0; r < 8; ++r)
            alpha[row0 + 8 * hi + r] = rd[r];
    }
}

// ---------------------------------------------------------------------------
__global__ void fill_u32(unsigned* __restrict__ p, unsigned val, int n) {
    int i = blockIdx.x * blockDim.x + threadIdx.x;
    if (i < n) p[i] = val;
}

// pass 1: segment max of leaky_relu(as[src] + ad[dst]) keyed by dst
__global__ void edge_max_kernel(const int* __restrict__ src,
                                const int* __restrict__ dst,
                                const float* __restrict__ as,
                                const float* __restrict__ ad,
                                unsigned* __restrict__ segmax, int E) {
    int i = blockIdx.x * blockDim.x + threadIdx.x;
    if (i >= E) return;
    const int s = src[i], d = dst[i];
    float e = as[s] + ad[d];
    e = (e > 0.f) ? e : NEG_SLOPE * e;
    atomicMax(&segmax[d], f2ord(e));
}

// pass 2 (fused): p = exp(e - m[dst]); z[dst] += p; accum[dst] += p*hs[src]
// C == 16 path: hs/accum rows are 64B-aligned -> float4 gathers (b128 loads).
template <int C>
__global__ void edge_scatter_kernel(const int* __restrict__ src,
                                    const int* __restrict__ dst,
                                    const float* __restrict__ as,
                                    const float* __restrict__ ad,
                                    const unsigned* __restrict__ segmax,
                                    const float* __restrict__ hs,
                                    float* __restrict__ z,
                                    float* __restrict__ accum, int E) {
    int i = blockIdx.x * blockDim.x + threadIdx.x;
    if (i >= E) return;
    const int s = src[i], d = dst[i];
    float e = as[s] + ad[d];
    e = (e > 0.f) ? e : NEG_SLOPE * e;
    const float m = ord2f(segmax[d]);
    const float p = expf(e - m);
    atomicAdd(&z[d], p);
    float* arow = accum + (size_t)d * C;
    if (C == 16) {
        const float4* hrow4 = (const float4*)(hs + (size_t)s * 16);
#pragma unroll
        for (int q = 0; q < 4; ++q) {
            const float4 h4 = hrow4[q];
            atomicAdd(&arow[4 * q + 0], p * h4.x);
            atomicAdd(&arow[4 * q + 1], p * h4.y);
            atomicAdd(&arow[4 * q + 2], p * h4.z);
            atomicAdd(&arow[4 * q + 3], p * h4.w);
        }
    } else {
        const float* hrow = hs + (size_t)s * C;
#pragma unroll
        for (int cc = 0; cc < C; ++cc)
            atomicAdd(&arow[cc], p * hrow[cc]);
    }
}

// out = accum/(z+1e-16) + bias  (optional ReLU)
__global__ void finalize_kernel(const float* __restrict__ accum,
                                const float* __restrict__ z,
                                const float* __restrict__ bias,
                                float* __restrict__ out,
                                int n, int C, int do_relu) {
    int i = blockIdx.x * blockDim.x + threadIdx.x;
    if (i >= n * C) return;
    const int d = i / C, cc = i - d * C;
    float v = accum[i] / (z[d] + 1e-16f) + bias[cc];
    if (do_relu) v = fmaxf(v, 0.f);
    out[i] = v;
}

// ---------------------------------------------------------------------------
static inline int cdiv(int a, int b) { return (a + b - 1) / b; }

extern "C" void kernel_launch(void* const* d_in, const int* in_sizes, int n_in,
                              void* d_out, int out_size, void* d_ws, size_t ws_size,
                              hipStream_t stream) {
    const float* x_user   = (const float*)d_in[0];
    const float* x_badge  = (const float*)d_in[1];
    /* edge_weight d_in[2] unused by reference */
    const float* W_src1   = (const float*)d_in[3];
    const float* W_dst1   = (const float*)d_in[4];
    const float* att_src1 = (const float*)d_in[5];
    const float* att_dst1 = (const float*)d_in[6];
    const float* bias1    = (const float*)d_in[7];
    const float* W_src2   = (const float*)d_in[8];
    const float* W_dst2   = (const float*)d_in[9];
    const float* att_src2 = (const float*)d_in[10];
    const float* att_dst2 = (const float*)d_in[11];
    const float* bias2    = (const float*)d_in[12];
    const int*   esrc     = (const int*)d_in[13];
    const int*   edst     = (const int*)d_in[14];

    const int NU = in_sizes[0] / 16;   // 200000
    const int NB = in_sizes[1] / 16;   // 50000
    const int E  = in_sizes[2];        // 5000000

    // ---- workspace carve-up (floats; every block is a multiple of 4 floats,
    //      so 16B alignment is preserved throughout) ----
    float* w = (float*)d_ws;
    size_t off = 0;
    float* hs1  = w + off; off += (size_t)NU * 16;
    float* hd1  = w + off; off += (size_t)NB * 16;
    float* as1  = w + off; off += (size_t)NU;
    float* ad1  = w + off; off += (size_t)NB;
    unsigned* sm1 = (unsigned*)(w + off); off += (size_t)NB;
    float* z1   = w + off; off += (size_t)NB;
    float* ac1  = w + off; off += (size_t)NB * 16;
    float* h1   = w + off; off += (size_t)NB * 16;
    float* hs2  = w + off; off += (size_t)NU * 2;
    float* hd2  = w + off; off += (size_t)NB * 2;
    float* as2  = w + off; off += (size_t)NU;
    float* ad2  = w + off; off += (size_t)NB;
    unsigned* sm2 = (unsigned*)(w + off); off += (size_t)NB;
    float* z2   = w + off; off += (size_t)NB;
    float* ac2  = w + off; off += (size_t)NB * 2;
    (void)ws_size;

    const int TB = 256;                 // 8 wave32s per block
    const unsigned ORD_NEG_INF = 0x007FFFFFu;  // f2ord(-inf)

    // ================= layer 1 =================
    gemm16_wmma<16><<<cdiv(NU / 16, 8), TB, 0, stream>>>(x_user, W_src1, att_src1,
                                                         hs1, as1, NU);
    gemm16_wmma<16><<<cdiv(NB / 16, 8), TB, 0, stream>>>(x_badge, W_dst1, att_dst1,
                                                         hd1, ad1, NB);
    fill_u32<<<cdiv(NB, TB), TB, 0, stream>>>(sm1, ORD_NEG_INF, NB);
    fill_u32<<<cdiv(NB, TB), TB, 0, stream>>>((unsigned*)z1, 0u, NB);
    fill_u32<<<cdiv(NB * 16, TB), TB, 0, stream>>>((unsigned*)ac1, 0u, NB * 16);

    edge_max_kernel<<<cdiv(E, TB), TB, 0, stream>>>(esrc, edst, as1, ad1, sm1, E);
    edge_scatter_kernel<16><<<cdiv(E, TB), TB, 0, stream>>>(esrc, edst, as1, ad1,
                                                            sm1, hs1, z1, ac1, E);
    finalize_kernel<<<cdiv(NB * 16, TB), TB, 0, stream>>>(ac1, z1, bias1, h1,
                                                          NB, 16, /*relu=*/1);

    // ================= layer 2 =================
    gemm16_wmma<2><<<cdiv(NU / 16, 8), TB, 0, stream>>>(x_user, W_src2, att_src2,
                                                        hs2, as2, NU);
    gemm16_wmma<2><<<cdiv(NB / 16, 8), TB, 0, stream>>>(h1, W_dst2, att_dst2,
                                                        hd2, ad2, NB);
    fill_u32<<<cdiv(NB, TB), TB, 0, stream>>>(sm2, ORD_NEG_INF, NB);
    fill_u32<<<cdiv(NB, TB), TB, 0, stream>>>((unsigned*)z2, 0u, NB);
    fill_u32<<<cdiv(NB * 2, TB), TB, 0, stream>>>((unsigned*)ac2, 0u, NB * 2);

    edge_max_kernel<<<cdiv(E, TB), TB, 0, stream>>>(esrc, edst, as2, ad2, sm2, E);
    edge_scatter_kernel<2><<<cdiv(E, TB), TB, 0, stream>>>(esrc, edst, as2, ad2,
                                                           sm2, hs2, z2, ac2, E);
    finalize_kernel<<<cdiv(NB * 2, TB), TB, 0, stream>>>(ac2, z2, bias2,
                                                         (float*)d_out,
                                                         NB, 2, /*relu=*/0);
}